// GreedyThresh_46076409152206
// MI455X (gfx1250) — compile-verified
//
#include <hip/hip_runtime.h>
#include <cstdint>

// ---------------------------------------------------------------------------
// Greedy-threshold sequential decode (B=4096 rows, V=128 steps, U1=129 cols).
// One wave32 per batch row. CDNA5 async global->LDS double-buffered prefetch
// (ASYNCcnt / s_wait_asynccnt) hides HBM latency of the per-step 129-float
// fetch while the wave does the masked-threshold argmax reduction in VALU/DS.
//
// Address trick: the async-load INST_OFFSET is added to BOTH the LDS address
// and the global address (CDNA5 ISA), and both sides place
// column c at byte 4c.  So one LDS VGPR (buf + 4*lane) and one global VGPR
// pair (step_base + 4*lane) serve all five transfers of a step via
// offset:0/128/256/384/512 immediates — no per-issue 64-bit adds.
// ---------------------------------------------------------------------------

#define DEC_V   128
#define DEC_U1  129
#define DEC_T   0.5f

#if defined(__has_builtin)
#  if __has_builtin(__builtin_amdgcn_s_wait_asynccnt)
#    define WAIT_ASYNC(n)                                        \
       do { __builtin_amdgcn_s_wait_asynccnt(n);                 \
            __asm__ volatile("" ::: "memory"); } while (0)
#  endif
#endif
#ifndef WAIT_ASYNC
#  define WAIT_ASYNC(n) __asm__ volatile("s_wait_asynccnt %0" :: "n"(n) : "memory")
#endif

// Issue one step's 129 floats into LDS: 4 full-wave dwords (cols 0..127,
// col = lane + 32k at offset 128k) plus a single-lane dword for col 128
// (same VGPR operands, offset:512, EXEC = lane 0 only).
// 5 ASYNCcnt increments per step regardless of EXEC.
__device__ __forceinline__ void async_step_issue(uint32_t lds_v, uint64_t ga,
                                                 int lane) {
  asm volatile(
      "global_load_async_to_lds_b32 %0, %1, off\n\t"
      "global_load_async_to_lds_b32 %0, %1, off offset:128\n\t"
      "global_load_async_to_lds_b32 %0, %1, off offset:256\n\t"
      "global_load_async_to_lds_b32 %0, %1, off offset:384"
      :: "v"(lds_v), "v"(ga) : "memory");
  if (lane == 0)
    asm volatile("global_load_async_to_lds_b32 %0, %1, off offset:512"
                 :: "v"(lds_v), "v"(ga) : "memory");
}

__device__ __forceinline__ uint64_t shflxor_u64(uint64_t k, int m) {
  uint32_t lo = __shfl_xor((uint32_t)k, m, 32);
  uint32_t hi = __shfl_xor((uint32_t)(k >> 32), m, 32);
  return ((uint64_t)hi << 32) | (uint64_t)lo;
}

// key = (float_bits(cand) << 32) | (511 - col): unsigned-max == larger value,
// ties -> smaller column index (matches jnp.argmax first-max semantics;
// candidates are always >= 0 so IEEE bits are order-preserving).
__device__ __forceinline__ uint64_t pack_key(float cand, int col) {
  return ((uint64_t)__float_as_uint(cand) << 32) | (uint32_t)(511 - col);
}

__global__ __launch_bounds__(32) void
GreedyThresh_46076409152206_kernel(const float* __restrict__ w,
                                   float* __restrict__ out, int B) {
  __shared__ __align__(16) float buf[2][132];   // cols 0..128 (+pad), x2 buffers
  const int lane = threadIdx.x;
  const int b = blockIdx.x;
  if (b >= B) return;

  const float* row = w + (size_t)b * DEC_V * DEC_U1;
  const uint32_t l0 = (uint32_t)(uintptr_t)&buf[0][0] + 4u * (uint32_t)lane;
  const uint32_t l1 = (uint32_t)(uintptr_t)&buf[1][0] + 4u * (uint32_t)lane;

  // Rolling prefetch pointer: per-lane global address of the NEXT step to issue.
  uint64_t ga = (uint64_t)(uintptr_t)row + 4ull * (uint32_t)lane;

  async_step_issue(l0, ga, lane); ga += 4ull * DEC_U1;   // step 0
  async_step_issue(l1, ga, lane); ga += 4ull * DEC_U1;   // step 1

  float    size   = 0.0f;  // accumulated chosen weights (replicated per lane)
  uint32_t dead   = 0;     // mask bits for this lane's cols 4*lane+j
  bool     m128   = false; // mask flag for col 128 (replicated)
  float    my_sel = 0.0f;  // staged sel for coalesced stores (lane i%32, step i)

  for (int i = 0; i < DEC_V; ++i) {
    // Double buffer: 5 async ops per step; keep step i+1's 5 in flight
    // except on the final iteration.
    if (i < DEC_V - 1) WAIT_ASYNC(5); else WAIT_ASYNC(0);

    float* bp = buf[i & 1];
    const float4 v   = *reinterpret_cast<const float4*>(bp + 4 * lane); // ds_load_b128
    const float  w0   = bp[0];    // original skip-slot weight (fallback "chosen")
    const float  x128 = bp[128];

    // Lane-local best over owned columns + (replicated) col 128.
    // Strict '>' keeps the earliest (lowest-col) candidate on exact ties.
    uint64_t k = 0;
    const float xs[4] = {v.x, v.y, v.z, v.w};
#pragma unroll
    for (int j = 0; j < 4; ++j) {
      const int   c    = 4 * lane + j;
      const bool  dd   = (dead >> j) & 1u;
      const float cand = (dd || xs[j] < DEC_T) ? 0.0f : xs[j];
      const uint64_t kj = pack_key(cand, c);
      if (kj > k) k = kj;
    }
    {
      const float cand = (m128 || x128 < DEC_T) ? 0.0f : x128;
      const uint64_t k8 = pack_key(cand, 128);
      if (k8 > k) k = k8;
    }

    // Wave32 butterfly max -> every lane holds the global best key.
#pragma unroll
    for (int off = 16; off >= 1; off >>= 1) {
      const uint64_t o = shflxor_u64(k, off);
      if (o > k) k = o;
    }

    const uint32_t vb = (uint32_t)(k >> 32);
    int   sel;
    float chosen;
    if (vb == 0u) {               // nothing passed mask+threshold -> forced skip
      sel = 0; chosen = w0;
    } else {
      sel = 511 - (int)(uint32_t)k;
      chosen = __uint_as_float(vb);
    }
    size += chosen;

    if (sel != 0) {               // skip column (col 0) is never masked
      if (sel == 128) m128 = true;
      else if ((sel >> 2) == lane) dead |= 1u << (sel & 3);
    }

    // Stage sel in lane (i%32); flush 32 at a time as coalesced b32 stores.
    if (lane == (i & 31)) my_sel = (float)sel;
    if ((i & 31) == 31)
      out[(size_t)B + (size_t)b * DEC_V + (size_t)(i & ~31) + lane] = my_sel;

    // Prefetch step i+2 into the buffer we just consumed.  All ds_load
    // results above are already consumed (hardware DScnt waits precede this
    // point), so overwriting the buffer via the async path is race-free.
    if (i + 2 < DEC_V) {
      async_step_issue((i & 1) ? l1 : l0, ga, lane);
      ga += 4ull * DEC_U1;
    }
  }

  if (lane == 0) out[b] = -size;   // reference returns -state.size
}

extern "C" void kernel_launch(void* const* d_in, const int* in_sizes, int n_in,
                              void* d_out, int out_size, void* d_ws, size_t ws_size,
                              hipStream_t stream) {
  (void)n_in; (void)out_size; (void)d_ws; (void)ws_size;
  const float* w = (const float*)d_in[0];
  float* out = (float*)d_out;
  const int B = in_sizes[0] / (DEC_V * DEC_U1);   // 4096 for the reference shapes
  GreedyThresh_46076409152206_kernel<<<dim3(B), dim3(32), 0, stream>>>(w, out, B);
}